// Inter_RM_6940667150684
// MI455X (gfx1250) — compile-verified
//
#include <hip/hip_runtime.h>
#include <hip/hip_bf16.h>
#include <math.h>

typedef __attribute__((ext_vector_type(2))) float v2f;
typedef __attribute__((ext_vector_type(8))) float v8f;

#define MNODE 9
#define DDIM  128
#define NTILE 16
// padded LDS strides for g to avoid 64-bank conflicts on 128-float rows
#define GSTR_M 132
#define GSTR_N (MNODE * GSTR_M)        // 1188 floats per example
#define G_FLOATS (NTILE * GSTR_N)      // 19008
#define E_OFF  G_FLOATS                // edges: 16 * 81
#define E_FLOATS (NTILE * MNODE * MNODE)
#define S_OFF  (E_OFF + E_FLOATS)      // s: 16 * 9
#define S_FLOATS (NTILE * MNODE)
#define SMEM_BYTES ((S_OFF + S_FLOATS) * sizeof(float))

__global__ void fused_graph_kernel(const float* __restrict__ f,
                                   const float* __restrict__ W,
                                   float* __restrict__ out) {
    extern __shared__ float smem[];
    float* g_lds = smem;           // [16][9][132]
    float* edges = smem + E_OFF;   // [16][9][9]
    float* ssum  = smem + S_OFF;   // [16][9]

    const int tid  = threadIdx.x;
    const int wave = tid >> 5;
    const int lane = tid & 31;
    const int l16  = lane & 15;
    const int half = lane >> 4;        // 0: lanes 0-15, 1: lanes 16-31
    const int n0   = blockIdx.x * NTILE;

    // zero the edges table (diagonal stays 0 => tanh(0) handled for free)
    for (int i = tid; i < E_FLOATS; i += blockDim.x) edges[i] = 0.0f;

    // ---------------- Phase 1: g = relu(f @ W^T) via fp32 WMMA ----------------
    // 72 output tiles = 9 slots (m) x 8 e-tiles of 16. Wave w takes t = w, w+8, ...
    for (int t = wave; t < MNODE * 8; t += 8) {
        const int m  = t >> 3;
        const int et = t & 7;

        // A fragment (16x4 fp32): lane row = l16, VGPR pair holds K = 2*half, 2*half+1
        const float* fa = f + (size_t)(n0 + l16) * (MNODE * DDIM) + m * DDIM + 2 * half;
        // B fragment (4x16 fp32): B[d][e] = W[m][e][d]; lane col = l16
        const float* wb = W + (size_t)m * (DDIM * DDIM) + (size_t)(et * 16 + l16) * DDIM + 2 * half;

        v8f c = {};
#pragma unroll 8
        for (int kk = 0; kk < DDIM / 4; ++kk) {
            v2f a = *(const v2f*)(fa + 4 * kk);
            v2f b = *(const v2f*)(wb + 4 * kk);
            c = __builtin_amdgcn_wmma_f32_16x16x4_f32(
                    /*neg_a=*/false, a, /*neg_b=*/false, b,
                    /*c_mod=*/(short)0, c, /*reuse_a=*/false, /*reuse_b=*/false);
        }

        // C layout: VGPR r -> row (r + 8*half), lane%16 -> col. ReLU, store to LDS.
#pragma unroll
        for (int r = 0; r < 8; ++r) {
            const int row = r + 8 * half;
            const float v = fmaxf(c[r], 0.0f);
            g_lds[row * GSTR_N + m * GSTR_M + et * 16 + l16] = v;
        }
    }
    __syncthreads();

    // ---------------- Phase 2a: edges[n,j,k] = tanh(||g_j - g_k||) ----------------
    // 36 unique pairs (j<k) per n, 16 n => 576 tasks
    for (int p = tid; p < NTILE * 36; p += blockDim.x) {
        const int n  = p / 36;
        int pp = p % 36;
        int j = 0;
        while (pp >= 8 - j) { pp -= (8 - j); ++j; }
        const int k = j + 1 + pp;

        const float* gj = g_lds + n * GSTR_N + j * GSTR_M;
        const float* gk = g_lds + n * GSTR_N + k * GSTR_M;
        float d2 = 0.0f;
#pragma unroll 4
        for (int e = 0; e < DDIM; ++e) {
            const float df = gj[e] - gk[e];
            d2 = fmaf(df, df, d2);
        }
        const float ed = (d2 > 0.0f) ? tanhf(sqrtf(d2)) : 0.0f;
        edges[n * (MNODE * MNODE) + j * MNODE + k] = ed;
        edges[n * (MNODE * MNODE) + k * MNODE + j] = ed;
    }
    __syncthreads();

    // ---------------- Phase 2b-pre: s[n,k] = sum_j edges[n,j,k] ----------------
    for (int t = tid; t < NTILE * MNODE; t += blockDim.x) {
        const int n = t / MNODE;
        const int k = t % MNODE;
        float acc = 0.0f;
#pragma unroll
        for (int j = 0; j < MNODE; ++j)
            acc += edges[n * (MNODE * MNODE) + j * MNODE + k];
        ssum[t] = acc;
    }
    __syncthreads();

    // ---------------- Phase 2c: y[n,e] = 0.5*sum_m f + 0.5*sum_k s[n,k]*g[n,k,e] ----------------
    for (int t = tid; t < NTILE * DDIM; t += blockDim.x) {
        const int n = t >> 7;
        const int e = t & (DDIM - 1);
        const float* fn = f + (size_t)(n0 + n) * (MNODE * DDIM) + e;
        float acc = 0.0f, fs = 0.0f;
#pragma unroll
        for (int k = 0; k < MNODE; ++k) {
            acc = fmaf(ssum[n * MNODE + k], g_lds[n * GSTR_N + k * GSTR_M + e], acc);
            fs += fn[k * DDIM];
        }
        out[(size_t)(n0 + n) * DDIM + e] = 0.5f * (fs + acc);
    }
}

extern "C" void kernel_launch(void* const* d_in, const int* in_sizes, int n_in,
                              void* d_out, int out_size, void* d_ws, size_t ws_size,
                              hipStream_t stream) {
    const float* f = (const float*)d_in[0];
    const float* W = (const float*)d_in[1];
    float* out = (float*)d_out;

    const int N = in_sizes[0] / (MNODE * DDIM);   // 8192
    const int nblocks = N / NTILE;                // 512

    // allow > 64 KB dynamic LDS (gfx1250 WGP has 320 KB)
    (void)hipFuncSetAttribute((const void*)fused_graph_kernel,
                              hipFuncAttributeMaxDynamicSharedMemorySize,
                              (int)SMEM_BYTES);

    fused_graph_kernel<<<nblocks, 256, SMEM_BYTES, stream>>>(f, W, out);
}